// TemporalCausalMLNN_30511447670811
// MI455X (gfx1250) — compile-verified
//
#include <hip/hip_runtime.h>
#include <hip/hip_bf16.h>
#include <math.h>

typedef __attribute__((ext_vector_type(2))) float v2f;
typedef __attribute__((ext_vector_type(8))) float v8f;

#define TAU_F 0.1f
#define WAVES_PER_BLOCK 8
#define BLOCK_A 256
#define GRID_A 1024

// Combine two online-softmax partial states (m = running max, z = sum exp,
// n = sum exp * coeff). m starts at -1e30f so exp() underflows to 0 instead
// of producing NaN for empty states.
__device__ __forceinline__ void online_combine(float& m, float& z, float& n,
                                               float m2, float z2, float n2) {
  float M = fmaxf(m, m2);
  float c1 = __expf(m - M);
  float c2 = __expf(m2 - M);
  z = z * c1 + z2 * c2;
  n = n * c1 + n2 * c2;
  m = M;
}

// fast sigmoid: v_exp_f32 + v_rcp_f32 (avoids the precise-division sequence)
__device__ __forceinline__ float fast_sigmoid(float x) {
  return __builtin_amdgcn_rcpf(1.0f + __expf(-x));
}

// Pass 1: fused gather + WMMA projection + online softmax partials.
// One wave processes 2x16 trace positions per iteration via two independent
// chains of 4x v_wmma_f32_16x16x4_f32 (K=4 chunks of the 16-wide embed dim),
// giving ~10 outstanding gathers per wave per iteration.
__global__ __launch_bounds__(BLOCK_A) void attn_partial_kernel(
    const int* __restrict__ trace, const float* __restrict__ embed,
    const float* __restrict__ w_attn, const float* __restrict__ b_attn,
    const float* __restrict__ causal, int seq_len,
    float* __restrict__ wsM, float* __restrict__ wsZ,
    float* __restrict__ wsN) {
  const int lane = threadIdx.x & 31;
  const int wave = threadIdx.x >> 5;
  const int hi   = lane >> 4;           // half-wave select
  const int numTiles = seq_len >> 4;    // 16 positions per tile
  const int gw = blockIdx.x * WAVES_PER_BLOCK + wave;
  const int gstride = gridDim.x * WAVES_PER_BLOCK;

  // B matrix (4x16 f32): w replicated into all 16 columns, so D[m,n] =
  // logit[m] for every n. Layout mirrors the documented f32 A layout:
  // VGPR0 = K{0 | 2}, VGPR1 = K{1 | 3} for lanes {0-15 | 16-31}.
  v2f B[4];
#pragma unroll
  for (int k = 0; k < 4; ++k) {
    B[k][0] = w_attn[4 * k + 2 * hi + 0];
    B[k][1] = w_attn[4 * k + 2 * hi + 1];
  }
  const float bias   = b_attn[0];
  const float invTau = 1.0f / TAU_F;

  const int r    = lane & 15;              // embed-tile row this lane loads
  const int own  = (lane & 7) + 8 * hi;    // logit row this lane owns (x2 dup,
                                           // cancels in N/Z)
  const int vsel = lane & 7;               // acc element holding row `own`

  float m = -1e30f, z = 0.0f, n = 0.0f;

  const int numPairs = numTiles >> 1;
  for (int p = gw; p < numPairs; p += gstride) {
    const int base0 = (p << 5);          // tile 2p,  positions base0..+15
    const int base1 = base0 + 16;        // tile 2p+1

    const int idx0 = trace[base0 + r];
    const int idx1 = trace[base1 + r];
    const float* row0 = embed + idx0 * 16 + 2 * hi;
    const float* row1 = embed + idx1 * 16 + 2 * hi;

    // issue all A-tile loads up front (compiler clauses them)
    v2f a0[4], a1[4];
#pragma unroll
    for (int k = 0; k < 4; ++k) {
      a0[k][0] = row0[4 * k + 0];
      a0[k][1] = row0[4 * k + 1];
      a1[k][0] = row1[4 * k + 0];
      a1[k][1] = row1[4 * k + 1];
    }

    // owned-position causality gathers (overlap with WMMA)
    const int iOwn0 = __shfl(idx0, own, 32);
    const int iOwn1 = __shfl(idx1, own, 32);
    const float sig0 = fast_sigmoid(causal[iOwn0]);
    const float sig1 = fast_sigmoid(causal[iOwn1]);

    v8f acc0 = {}, acc1 = {};
#pragma unroll
    for (int k = 0; k < 4; ++k) {
      acc0 = __builtin_amdgcn_wmma_f32_16x16x4_f32(
          false, a0[k], false, B[k], (short)0, acc0, false, false);
      acc1 = __builtin_amdgcn_wmma_f32_16x16x4_f32(
          false, a1[k], false, B[k], (short)0, acc1, false, false);
    }

    // extract this lane's owned logits (row `own` sits in acc[lane&7])
    float raw0 = acc0[0], raw1 = acc1[0];
#pragma unroll
    for (int t = 1; t < 8; ++t) {
      raw0 = (vsel == t) ? acc0[t] : raw0;
      raw1 = (vsel == t) ? acc1[t] : raw1;
    }
    const float s0 = (raw0 + bias) * invTau;
    const float s1 = (raw1 + bias) * invTau;

    // online softmax update (combine the pair first, then fold into state)
    const float Mp  = fmaxf(s0, s1);
    const float e0  = __expf(s0 - Mp);
    const float e1  = __expf(s1 - Mp);
    const float M   = fmaxf(m, Mp);
    const float cm  = __expf(m - M);
    const float cp  = __expf(Mp - M);
    z = z * cm + (e0 + e1) * cp;
    n = n * cm + (e0 * sig0 + e1 * sig1) * cp;
    m = M;
  }

  // remainder tile (wave-uniform branch; EXEC stays all-1s)
  if ((numTiles & 1) && gw == 0) {
    const int base = (numTiles - 1) << 4;
    const int idx  = trace[base + r];
    const float* row = embed + idx * 16 + 2 * hi;
    v8f acc = {};
#pragma unroll
    for (int k = 0; k < 4; ++k) {
      v2f a;
      a[0] = row[4 * k + 0];
      a[1] = row[4 * k + 1];
      acc = __builtin_amdgcn_wmma_f32_16x16x4_f32(
          false, a, false, B[k], (short)0, acc, false, false);
    }
    float raw = acc[0];
#pragma unroll
    for (int t = 1; t < 8; ++t) raw = (vsel == t) ? acc[t] : raw;
    const float s   = (raw + bias) * invTau;
    const int iOwn  = __shfl(idx, own, 32);
    const float sig = fast_sigmoid(causal[iOwn]);
    const float M   = fmaxf(m, s);
    const float cm  = __expf(m - M);
    const float e   = __expf(s - M);
    z = z * cm + e;
    n = n * cm + e * sig;
    m = M;
  }

  // wave32 butterfly reduction of (m, z, n)
#pragma unroll
  for (int off = 16; off >= 1; off >>= 1) {
    float m2 = __shfl_xor(m, off, 32);
    float z2 = __shfl_xor(z, off, 32);
    float n2 = __shfl_xor(n, off, 32);
    online_combine(m, z, n, m2, z2, n2);
  }

  __shared__ float sm[WAVES_PER_BLOCK], sz[WAVES_PER_BLOCK],
      sn[WAVES_PER_BLOCK];
  if (lane == 0) { sm[wave] = m; sz[wave] = z; sn[wave] = n; }
  __syncthreads();
  if (threadIdx.x == 0) {
    float bm = sm[0], bz = sz[0], bn = sn[0];
    for (int i = 1; i < WAVES_PER_BLOCK; ++i)
      online_combine(bm, bz, bn, sm[i], sz[i], sn[i]);
    wsM[blockIdx.x] = bm;
    wsZ[blockIdx.x] = bz;
    wsN[blockIdx.x] = bn;
  }
}

// Pass 2: single block combines the per-block partials and finalizes.
__global__ __launch_bounds__(256) void finalize_kernel(
    const float* __restrict__ wsM, const float* __restrict__ wsZ,
    const float* __restrict__ wsN, int nparts,
    const int* __restrict__ is_crash, float* __restrict__ out) {
  const int lane = threadIdx.x & 31;
  const int wave = threadIdx.x >> 5;

  float m = -1e30f, z = 0.0f, n = 0.0f;
  for (int i = threadIdx.x; i < nparts; i += blockDim.x)
    online_combine(m, z, n, wsM[i], wsZ[i], wsN[i]);

#pragma unroll
  for (int off = 16; off >= 1; off >>= 1) {
    float m2 = __shfl_xor(m, off, 32);
    float z2 = __shfl_xor(z, off, 32);
    float n2 = __shfl_xor(n, off, 32);
    online_combine(m, z, n, m2, z2, n2);
  }

  __shared__ float sm[8], sz[8], sn[8];
  if (lane == 0) { sm[wave] = m; sz[wave] = z; sn[wave] = n; }
  __syncthreads();
  if (threadIdx.x == 0) {
    float bm = sm[0], bz = sz[0], bn = sn[0];
    for (int i = 1; i < 8; ++i) online_combine(bm, bz, bn, sm[i], sz[i], sn[i]);
    float explained = (bz > 0.0f) ? (bn / bz) : 0.0f;  // precise div, once
    explained = fminf(fmaxf(explained, 0.0f), 1.0f);
    const float crash = (float)is_crash[0];
    out[0] = (crash > 0.5f) ? fmaxf(1.0f - explained, 0.0f)
                            : fmaxf(explained, 0.0f);
  }
}

extern "C" void kernel_launch(void* const* d_in, const int* in_sizes, int n_in,
                              void* d_out, int out_size, void* d_ws,
                              size_t ws_size, hipStream_t stream) {
  const int*   trace    = (const int*)d_in[0];
  const int*   is_crash = (const int*)d_in[1];
  const float* embed    = (const float*)d_in[2];
  const float* w_attn   = (const float*)d_in[3];
  const float* b_attn   = (const float*)d_in[4];
  const float* causal   = (const float*)d_in[5];
  const int seq_len = in_sizes[0];

  int grid = GRID_A;
  const size_t need = (size_t)grid * 3u * sizeof(float);
  if (need > ws_size && ws_size >= 3 * sizeof(float))
    grid = (int)(ws_size / (3 * sizeof(float)));
  if (grid < 1) grid = 1;

  float* wsf = (float*)d_ws;
  float* wsM = wsf;
  float* wsZ = wsf + grid;
  float* wsN = wsf + 2 * grid;

  attn_partial_kernel<<<grid, BLOCK_A, 0, stream>>>(
      trace, embed, w_attn, b_attn, causal, seq_len, wsM, wsZ, wsN);
  finalize_kernel<<<1, 256, 0, stream>>>(wsM, wsZ, wsN, grid, is_crash,
                                         (float*)d_out);
}